// HeteroGAT_5325759447206
// MI455X (gfx1250) — compile-verified
//
#include <hip/hip_runtime.h>
#include <hip/hip_bf16.h>
#include <math.h>

typedef __attribute__((ext_vector_type(2))) float v2f;
typedef __attribute__((ext_vector_type(8))) float v8f;

#define DIM 64
#define NEG_SLOPE 0.2f

__device__ __forceinline__ float lrelu(float v) { return v > 0.f ? v : NEG_SLOPE * v; }

// order-preserving float <-> uint encoding for atomicMax on floats
__device__ __forceinline__ unsigned int fenc(float f) {
    unsigned int u = __float_as_uint(f);
    return (u & 0x80000000u) ? ~u : (u | 0x80000000u);
}
__device__ __forceinline__ float fdec(unsigned int u) {
    return __uint_as_float((u & 0x80000000u) ? (u & 0x7FFFFFFFu) : ~u);
}

// ---------------------------------------------------------------------------
// K1: h_login = x @ W_login ; h_exec = x @ W_exec via V_WMMA_F32_16X16X4_F32.
// One wave per 16-row tile; 4 col-tiles x 2 branches = 8 accumulators.
// A fragment layout (16x4 f32): lanes 0-15 hold row M=lane, K={0,1}; lanes
// 16-31 hold same rows, K={2,3}. B fragment mirrors with N=lane&15.
// ---------------------------------------------------------------------------
__global__ __launch_bounds__(256) void k_gemm(
    const float* __restrict__ x, const float* __restrict__ Wl,
    const float* __restrict__ We, float* __restrict__ hl,
    float* __restrict__ he, int N)
{
    int wave  = blockIdx.x * (blockDim.x >> 5) + (threadIdx.x >> 5);
    int lane  = threadIdx.x & 31;
    int row0  = wave * 16;
    if (row0 >= N) return;                    // wave-uniform: EXEC stays full

    int mlane = lane & 15;                    // A: row-in-tile; B/C: col-in-tile
    int khalf = (lane >> 4) << 1;             // 0 or 2

    int  arow     = row0 + mlane;
    bool rowValid = arow < N;
    const float* aptr = x + (size_t)(rowValid ? arow : 0) * DIM;

    v8f accL[4] = {}, accE[4] = {};

    for (int k = 0; k < DIM; k += 4) {
        v2f a;
        a.x = aptr[k + khalf + 0];
        a.y = aptr[k + khalf + 1];
        if (!rowValid) { a.x = 0.f; a.y = 0.f; }   // lane-level select, no EXEC change
#pragma unroll
        for (int t = 0; t < 4; ++t) {
            int col = t * 16 + mlane;
            v2f bl, be;
            bl.x = Wl[(k + khalf + 0) * DIM + col];
            bl.y = Wl[(k + khalf + 1) * DIM + col];
            be.x = We[(k + khalf + 0) * DIM + col];
            be.y = We[(k + khalf + 1) * DIM + col];
            accL[t] = __builtin_amdgcn_wmma_f32_16x16x4_f32(
                false, a, false, bl, (short)0, accL[t], false, false);
            accE[t] = __builtin_amdgcn_wmma_f32_16x16x4_f32(
                false, a, false, be, (short)0, accE[t], false, false);
        }
    }

    // C/D layout: lanes 0-15 VGPR r -> M=r; lanes 16-31 VGPR r -> M=8+r; N=lane&15
    int rbase = row0 + ((lane >> 4) ? 8 : 0);
#pragma unroll
    for (int r = 0; r < 8; ++r) {
        int rr = rbase + r;
        if (rr < N) {
            size_t off = (size_t)rr * DIM + mlane;
#pragma unroll
            for (int t = 0; t < 4; ++t) {
                hl[off + t * 16] = accL[t][r];
                he[off + t * 16] = accE[t][r];
            }
        }
    }
}

// ---------------------------------------------------------------------------
// K2: s = h @ a_src, d = h @ a_dst. 16 lanes per row, float4 per lane.
// ---------------------------------------------------------------------------
__global__ __launch_bounds__(256) void k_sd(
    const float* __restrict__ h, const float* __restrict__ a_s,
    const float* __restrict__ a_d, float* __restrict__ s,
    float* __restrict__ d, int N)
{
    int tid = blockIdx.x * blockDim.x + threadIdx.x;
    int row = tid >> 4;
    int l16 = tid & 15;
    if (row >= N) return;
    float4 hv  = ((const float4*)(h + (size_t)row * DIM))[l16];
    float4 as4 = ((const float4*)a_s)[l16];
    float4 ad4 = ((const float4*)a_d)[l16];
    float ps = hv.x * as4.x + hv.y * as4.y + hv.z * as4.z + hv.w * as4.w;
    float pd = hv.x * ad4.x + hv.y * ad4.y + hv.z * ad4.z + hv.w * ad4.w;
#pragma unroll
    for (int off = 8; off; off >>= 1) {
        ps += __shfl_xor(ps, off, 16);
        pd += __shfl_xor(pd, off, 16);
    }
    if (l16 == 0) { s[row] = ps; d[row] = pd; }
}

// K3: init segment-max with the self-loop term (always present).
__global__ __launch_bounds__(256) void k_minit(
    const float* sl, const float* dl, const float* se, const float* de,
    unsigned int* ml, unsigned int* me, int N)
{
    int i = blockIdx.x * blockDim.x + threadIdx.x;
    if (i >= N) return;
    ml[i] = fenc(lrelu(sl[i] + dl[i]));
    me[i] = fenc(lrelu(se[i] + de[i]));
}

// K4: per-edge atomic max into the edge's own branch (masked edges are no-ops).
__global__ __launch_bounds__(256) void k_emax(
    const int* __restrict__ src, const int* __restrict__ dst,
    const int* __restrict__ typ, const float* sl, const float* dl,
    const float* se, const float* de, unsigned int* ml, unsigned int* me, int E)
{
    int e = blockIdx.x * blockDim.x + threadIdx.x;
    if (e >= E) return;
    int u = src[e], v = dst[e], t = typ[e];
    float ev = t ? lrelu(se[u] + de[v]) : lrelu(sl[u] + dl[v]);
    atomicMax(t ? (me + v) : (ml + v), fenc(ev));
}

// K5: denom initialized with self-loop contribution.
__global__ __launch_bounds__(256) void k_dself(
    const float* sl, const float* dl, const float* se, const float* de,
    const unsigned int* ml, const unsigned int* me,
    float* denl, float* dene, int N)
{
    int i = blockIdx.x * blockDim.x + threadIdx.x;
    if (i >= N) return;
    denl[i] = expf(lrelu(sl[i] + dl[i]) - fdec(ml[i]));
    dene[i] = expf(lrelu(se[i] + de[i]) - fdec(me[i]));
}

// K6: per-edge exp-sum into denom.
__global__ __launch_bounds__(256) void k_eden(
    const int* __restrict__ src, const int* __restrict__ dst,
    const int* __restrict__ typ, const float* sl, const float* dl,
    const float* se, const float* de, const unsigned int* ml,
    const unsigned int* me, float* denl, float* dene, int E)
{
    int e = blockIdx.x * blockDim.x + threadIdx.x;
    if (e >= E) return;
    int u = src[e], v = dst[e], t = typ[e];
    float ev = t ? lrelu(se[u] + de[v]) : lrelu(sl[u] + dl[v]);
    float p  = expf(ev - fdec(t ? me[v] : ml[v]));
    atomicAdd(t ? (dene + v) : (denl + v), p);
}

// K7: wave-per-edge message scatter: out[dst] += alpha * h[src].
__global__ __launch_bounds__(256) void k_eacc(
    const int* __restrict__ src, const int* __restrict__ dst,
    const int* __restrict__ typ, const float* __restrict__ hl,
    const float* __restrict__ he, const float* sl, const float* dl,
    const float* se, const float* de, const unsigned int* ml,
    const unsigned int* me, const float* denl, const float* dene,
    float* __restrict__ out, int E)
{
    int wid  = (blockIdx.x * blockDim.x + threadIdx.x) >> 5;
    int lane = threadIdx.x & 31;
    if (wid >= E) return;
    int u = src[wid], v = dst[wid], t = typ[wid];
    float ev    = t ? lrelu(se[u] + de[v]) : lrelu(sl[u] + dl[v]);
    float alpha = expf(ev - fdec(t ? me[v] : ml[v])) / (t ? dene[v] : denl[v]);
    const float* H = t ? he : hl;
    float2 hv = ((const float2*)(H + (size_t)u * DIM))[lane];
    float* o  = out + (size_t)v * DIM + lane * 2;
    atomicAdd(o + 0, hv.x * alpha);
    atomicAdd(o + 1, hv.y * alpha);
}

// K8: wave-per-node: add self-loop terms of both branches plus biases.
__global__ __launch_bounds__(256) void k_final(
    const float* __restrict__ hl, const float* __restrict__ he,
    const float* sl, const float* dl, const float* se, const float* de,
    const unsigned int* ml, const unsigned int* me,
    const float* denl, const float* dene,
    const float* __restrict__ bl, const float* __restrict__ be,
    float* __restrict__ out, int N)
{
    int w    = (blockIdx.x * blockDim.x + threadIdx.x) >> 5;
    int lane = threadIdx.x & 31;
    if (w >= N) return;
    float wl_ = expf(lrelu(sl[w] + dl[w]) - fdec(ml[w])) / denl[w];
    float we_ = expf(lrelu(se[w] + de[w]) - fdec(me[w])) / dene[w];
    float2 a   = ((const float2*)(hl + (size_t)w * DIM))[lane];
    float2 b2  = ((const float2*)(he + (size_t)w * DIM))[lane];
    float2 bl2 = ((const float2*)bl)[lane];
    float2 be2 = ((const float2*)be)[lane];
    size_t o = (size_t)w * DIM + lane * 2;
    out[o + 0] += a.x * wl_ + b2.x * we_ + bl2.x + be2.x;
    out[o + 1] += a.y * wl_ + b2.y * we_ + bl2.y + be2.y;
}

extern "C" void kernel_launch(void* const* d_in, const int* in_sizes, int n_in,
                              void* d_out, int out_size, void* d_ws, size_t ws_size,
                              hipStream_t stream)
{
    const float* x    = (const float*)d_in[0];
    const int*   eidx = (const int*)d_in[1];
    const int*   etyp = (const int*)d_in[2];
    const float* Wl   = (const float*)d_in[3];
    const float* asl  = (const float*)d_in[4];
    const float* adl  = (const float*)d_in[5];
    const float* bl   = (const float*)d_in[6];
    const float* We   = (const float*)d_in[7];
    const float* ase  = (const float*)d_in[8];
    const float* ade  = (const float*)d_in[9];
    const float* be   = (const float*)d_in[10];

    const int N = in_sizes[0] / DIM;
    const int E = in_sizes[2];
    const int* src = eidx;
    const int* dst = eidx + E;

    // workspace carve-up (all f32/u32): 2*64N + 8N floats ~= 54.4 MB
    float* ws = (float*)d_ws;
    float* hl = ws;
    float* he = hl + (size_t)N * DIM;
    float* sl = he + (size_t)N * DIM;
    float* dl = sl + N;
    float* se = dl + N;
    float* de = se + N;
    unsigned int* ml = (unsigned int*)(de + N);
    unsigned int* me = ml + N;
    float* denl = (float*)(me + N);
    float* dene = denl + N;

    float* out = (float*)d_out;
    hipMemsetAsync(out, 0, (size_t)N * DIM * sizeof(float), stream);

    const int TB = 256;
    int tilesN   = (N + 15) / 16;
    int gemmBlk  = (tilesN + 7) / 8;                 // 8 waves / block
    k_gemm<<<gemmBlk, TB, 0, stream>>>(x, Wl, We, hl, he, N);

    int sdBlk = ((size_t)N * 16 + TB - 1) / TB;
    k_sd<<<sdBlk, TB, 0, stream>>>(hl, asl, adl, sl, dl, N);
    k_sd<<<sdBlk, TB, 0, stream>>>(he, ase, ade, se, de, N);

    int nBlk = (N + TB - 1) / TB;
    int eBlk = (E + TB - 1) / TB;
    k_minit<<<nBlk, TB, 0, stream>>>(sl, dl, se, de, ml, me, N);
    k_emax <<<eBlk, TB, 0, stream>>>(src, dst, etyp, sl, dl, se, de, ml, me, E);
    k_dself<<<nBlk, TB, 0, stream>>>(sl, dl, se, de, ml, me, denl, dene, N);
    k_eden <<<eBlk, TB, 0, stream>>>(src, dst, etyp, sl, dl, se, de, ml, me,
                                     denl, dene, E);

    int eaccBlk = ((size_t)E * 32 + TB - 1) / TB;    // wave per edge
    k_eacc<<<eaccBlk, TB, 0, stream>>>(src, dst, etyp, hl, he, sl, dl, se, de,
                                       ml, me, denl, dene, out, E);

    int finBlk = ((size_t)N * 32 + TB - 1) / TB;     // wave per node
    k_final<<<finBlk, TB, 0, stream>>>(hl, he, sl, dl, se, de, ml, me,
                                       denl, dene, bl, be, out, N);
}